// SharpeningKernel_54932631715899
// MI455X (gfx1250) — compile-verified
//
#include <hip/hip_runtime.h>
#include <math.h>

// ---------------------------------------------------------------------------
// PGD force kernel for MI455X (gfx1250, wave32, WMMA + TDM).
// 3D FFTs as DFT-GEMMs on the f32 matrix pipe (V_WMMA_F32_16X16X4_F32),
// axis-rotating each pass. k-space filter fused into the last forward pass,
// gradient (-i*k) fused into the first inverse pass. Twiddle B-tiles staged
// in LDS via the Tensor Data Mover (tensor_load_to_lds + s_wait_tensorcnt).
// ---------------------------------------------------------------------------

typedef float v2f __attribute__((ext_vector_type(2)));
typedef float v8f __attribute__((ext_vector_type(8)));
typedef unsigned int u32x4 __attribute__((ext_vector_type(4)));
typedef int i32x4 __attribute__((ext_vector_type(4)));
typedef int i32x8 __attribute__((ext_vector_type(8)));

#define MESH   128
#define MESH3  (MESH * MESH * MESH)
#define NLINES (MESH * MESH)          // 16384 lines per axis pass
#define ALPHA  0.2f
#define KL     0.3f
#define KS     1.0f
#define TWO_PI 6.283185307179586f

#define PGD_USE_TDM 1

__device__ __forceinline__ float kfreq(int i) {
    int s = (i < MESH / 2) ? i : i - MESH;
    return TWO_PI * (float)s / (float)MESH;
}

// ---------------------------------------------------------------------------
__global__ void pgd_zero(float* __restrict__ p, int n) {
    int i = blockIdx.x * blockDim.x + threadIdx.x;
    if (i < n) p[i] = 0.0f;
}

// Symmetric DFT twiddles M[k][j] = exp(-i*2pi*k*j/128); exact mod-128 angle
// reduction. Inverse matrix carries 1/128 (1/128^3 total over 3 passes).
__global__ void pgd_twiddles(float* __restrict__ fr, float* __restrict__ fi,
                             float* __restrict__ ir, float* __restrict__ ii) {
    int t = blockIdx.x * blockDim.x + threadIdx.x;
    if (t >= NLINES) return;
    int k = t >> 7, j = t & (MESH - 1);
    int m = (k * j) & (MESH - 1);
    float a = -TWO_PI * (float)m / (float)MESH;
    float c = cosf(a), s = sinf(a);
    fr[t] = c;
    fi[t] = s;
    ir[t] = c * (1.0f / (float)MESH);
    ii[t] = -s * (1.0f / (float)MESH);
}

// ---------------------------------------------------------------------------
// CIC paint: 8 f32 atomics per particle into the 8MB L2-resident grid.
__global__ void pgd_paint(const float* __restrict__ pos,
                          float* __restrict__ grid, int n) {
    int i = blockIdx.x * blockDim.x + threadIdx.x;
    if (i >= n) return;
    float px = pos[3 * i + 0], py = pos[3 * i + 1], pz = pos[3 * i + 2];
    float fx = floorf(px), fy = floorf(py), fz = floorf(pz);
    int ix = (int)fx & (MESH - 1), iy = (int)fy & (MESH - 1), iz = (int)fz & (MESH - 1);
    float dx = px - fx, dy = py - fy, dz = pz - fz;
    float wx[2] = {1.0f - dx, dx}, wy[2] = {1.0f - dy, dy}, wz[2] = {1.0f - dz, dz};
#pragma unroll
    for (int a = 0; a < 2; ++a)
#pragma unroll
        for (int b = 0; b < 2; ++b)
#pragma unroll
            for (int c = 0; c < 2; ++c) {
                int cx = (ix + a) & (MESH - 1);
                int cy = (iy + b) & (MESH - 1);
                int cz = (iz + c) & (MESH - 1);
                atomicAdd(&grid[((cx * MESH) + cy) * MESH + cz],
                          wx[a] * wy[b] * wz[c]);
            }
}

// ---------------------------------------------------------------------------
#if PGD_USE_TDM && __has_builtin(__builtin_amdgcn_tensor_load_to_lds)
// TDM descriptor per CDNA5 ISA:
// 2D tile: 128 rows x 16 elems, 4B elements, tensor row stride 128 elems.
// This toolchain exposes the 6-arg builtin form:
//   (u32x4 g0, i32x8 g1, i32x4 g2, i32x4 g3, i32x8 pad, i32 cpol)
__device__ __forceinline__ void tdm_load_tile(unsigned ldsByteOff,
                                              const float* gsrc) {
    unsigned long long ga = (unsigned long long)(size_t)gsrc;
    u32x4 g0;
    g0.x = 1u;                                   // count=1, user desc, no gather
    g0.y = ldsByteOff;                           // lds_addr (bytes)
    g0.z = (unsigned)(ga & 0xffffffffu);         // global_addr[31:0]
    g0.w = (unsigned)((ga >> 32) & 0x1ffffffu)   // global_addr[56:32]
           | (2u << 30);                         // type = 2 ("image")
    i32x8 g1;
    g1[0] = (2 << 16);                           // data_size=2 (4B), wg_mask=0
    g1[1] = (MESH << 16);                        // tensor_dim0[15:0] = 128
    g1[2] = (MESH << 16);                        // tensor_dim1[15:0] = 128
    g1[3] = (16 << 16);                          // tile_dim0 = 16
    g1[4] = MESH;                                // tile_dim1 = 128
    g1[5] = MESH;                                // tensor_dim0_stride = 128
    g1[6] = 0;                                   // (dim1 stride unused for 2D)
    g1[7] = 0;
    i32x4 gz = {0, 0, 0, 0};                     // groups 2/3 unused (2D)
    i32x8 gz8 = {0, 0, 0, 0, 0, 0, 0, 0};
    __builtin_amdgcn_tensor_load_to_lds(g0, g1, gz, gz, gz8, 0);
}
#endif

// ---------------------------------------------------------------------------
// One axis pass of the 3D (i)FFT as a complex GEMM on V_WMMA_F32_16X16X4_F32.
//   in  : [d0][d1][128]  (contiguous transform axis, line = d0*d1 index)
//   out : [128][d0*d1]   (transformed axis rotated to outermost)
// Block = 8 waves: one 16-column tile (staged in LDS once) x 8 line tiles.
// FILT: fuse PGD band-pass * inverse Laplacian at the store (fwd pass 3,
//       output layout [kx][ky][kz], kx = jb+r per lane).
// GRAD: fuse g = -i*k_axis*pot at the load (inv pass 1, input [kx][ky][kz]).
// Fragment layouts per ISA 7.12.2; f32 WMMA has no A/B NEG, so the -Im*Im
// term negates the 2-element A fragment in VALU.
template <bool HAS_IM, bool WRITE_IM, bool FILT, bool GRAD>
__global__ __launch_bounds__(256) void pgd_fft_pass(
    const float* __restrict__ inRe, const float* __restrict__ inIm,
    const float* __restrict__ mRe, const float* __restrict__ mIm,
    float* __restrict__ outRe, float* __restrict__ outIm, int gradAxis) {
    __shared__ float sBre[MESH][16];   // [row k][col] : conflict-free reads
    __shared__ float sBim[MESH][16];

    const int jb = (blockIdx.x & 7) << 4;          // column tile of 16

#if PGD_USE_TDM && __has_builtin(__builtin_amdgcn_tensor_load_to_lds)
    if (threadIdx.x < 32) {                        // wave 0 drives the TDM
        tdm_load_tile((unsigned)(size_t)(void*)&sBre[0][0], mRe + jb);
        tdm_load_tile((unsigned)(size_t)(void*)&sBim[0][0], mIm + jb);
#if __has_builtin(__builtin_amdgcn_s_wait_tensorcnt)
        __builtin_amdgcn_s_wait_tensorcnt(0);
#endif
    }
#else
    for (int t = threadIdx.x; t < MESH * 16; t += 256) {
        int row = t >> 4, col = t & 15;
        sBre[row][col] = mRe[row * MESH + jb + col];
        sBim[row][col] = mIm[row * MESH + jb + col];
    }
#endif
    __syncthreads();

    const int lane     = threadIdx.x & 31;
    const int lineTile = ((blockIdx.x >> 3) << 3) + (threadIdx.x >> 5);
    const int lineBase = lineTile << 4;
    const int r        = lane & 15;
    const bool hi      = lane >= 16;
    const int koff     = hi ? 2 : 0;
    const int line     = lineBase + r;

    float kfRow = 0.0f;                 // per-lane-constant gradient wavenumber
    if (GRAD) {
        if (gradAxis == 0)      kfRow = kfreq(line >> 7);
        else if (gradAxis == 1) kfRow = kfreq(line & (MESH - 1));
    }

    const float* rowRe = inRe + (size_t)line * MESH;
    const float* rowIm = HAS_IM ? (inIm + (size_t)line * MESH) : nullptr;

    v8f accRe = {};
    v8f accIm = {};

#pragma unroll 4
    for (int k = 0; k < MESH; k += 4) {
        v2f aRe = *(const v2f*)(rowRe + k + koff);       // one b64 load
        v2f aIm;
        if (HAS_IM) aIm = *(const v2f*)(rowIm + k + koff);
        else { aIm.x = 0.0f; aIm.y = 0.0f; }

        if (GRAD) {                                      // (gRe,gIm)=(k*pi,-k*pr)
            v2f kf;
            if (gradAxis == 2) { kf.x = kfreq(k + koff); kf.y = kfreq(k + koff + 1); }
            else               { kf.x = kfRow;           kf.y = kfRow; }
            v2f gRe = kf * aIm;
            v2f gIm = -kf * aRe;
            aRe = gRe;
            aIm = gIm;
        }

        v2f bRe, bIm;                                    // ds loads from LDS tile
        bRe.x = sBre[k + koff + 0][r];
        bRe.y = sBre[k + koff + 1][r];
        bIm.x = sBim[k + koff + 0][r];
        bIm.y = sBim[k + koff + 1][r];

        accRe = __builtin_amdgcn_wmma_f32_16x16x4_f32(false, aRe, false, bRe,
                                                      (short)0, accRe, false, false);
        accIm = __builtin_amdgcn_wmma_f32_16x16x4_f32(false, aRe, false, bIm,
                                                      (short)0, accIm, false, false);
        if (HAS_IM || GRAD) {
            v2f aImN = -aIm;
            accRe = __builtin_amdgcn_wmma_f32_16x16x4_f32(false, aImN, false, bIm,
                                                          (short)0, accRe, false, false);
            accIm = __builtin_amdgcn_wmma_f32_16x16x4_f32(false, aIm, false, bRe,
                                                          (short)0, accIm, false, false);
        }
    }

    const int mBase = hi ? 8 : 0;

    if (FILT) {                      // out layout [kx][ky][kz]; kx = jb+r
        float fx = kfreq(jb + r);
#pragma unroll
        for (int e = 0; e < 8; ++e) {
            int l2 = lineBase + mBase + e;
            float fy = kfreq(l2 >> 7);
            float fz = kfreq(l2 & (MESH - 1));
            float kk = fx * fx + fy * fy + fz * fz;
            float s = 0.0f;
            if (kk != 0.0f) {
                s = -(1.0f / kk) * __expf(-(KL * KL) / kk)
                    * __expf(-(kk * kk) / (KS * KS * KS * KS));
            }
            accRe[e] *= s;
            accIm[e] *= s;
        }
    }

    // Rotated store: out[(jb+n)*NLINES + line]; 32B per lane, 32B aligned.
    size_t ob = (size_t)(jb + r) * NLINES + (size_t)(lineBase + mBase);
    *(v8f*)(outRe + ob) = accRe;
    if (WRITE_IM) *(v8f*)(outIm + ob) = accIm;
}

// ---------------------------------------------------------------------------
// CIC gather of one force component + velocity boost for that component.
__global__ void pgd_read(const float* __restrict__ f, const float* __restrict__ pos,
                         const float* __restrict__ vel, const float* __restrict__ fdf,
                         float* __restrict__ outVel, int axis, int n) {
    int i = blockIdx.x * blockDim.x + threadIdx.x;
    if (i >= n) return;
    float px = pos[3 * i + 0], py = pos[3 * i + 1], pz = pos[3 * i + 2];
    float fx = floorf(px), fy = floorf(py), fz = floorf(pz);
    int ix = (int)fx & (MESH - 1), iy = (int)fy & (MESH - 1), iz = (int)fz & (MESH - 1);
    float dx = px - fx, dy = py - fy, dz = pz - fz;
    float wx[2] = {1.0f - dx, dx}, wy[2] = {1.0f - dy, dy}, wz[2] = {1.0f - dz, dz};
    float acc = 0.0f;
#pragma unroll
    for (int a = 0; a < 2; ++a)
#pragma unroll
        for (int b = 0; b < 2; ++b)
#pragma unroll
            for (int c = 0; c < 2; ++c) {
                int cx = (ix + a) & (MESH - 1);
                int cy = (iy + b) & (MESH - 1);
                int cz = (iz + c) & (MESH - 1);
                acc += f[((cx * MESH) + cy) * MESH + cz] * wx[a] * wy[b] * wz[c];
            }
    outVel[3 * i + axis] = vel[3 * i + axis] + (acc * ALPHA) / fdf[0];
}

// ---------------------------------------------------------------------------
extern "C" void kernel_launch(void* const* d_in, const int* in_sizes, int n_in,
                              void* d_out, int out_size, void* d_ws, size_t ws_size,
                              hipStream_t stream) {
    const float* pos = (const float*)d_in[0];
    const float* vel = (const float*)d_in[1];
    const float* fdf = (const float*)d_in[2];
    const int n = in_sizes[0] / 3;         // 128^3 particles

    float* outPos = (float*)d_out;
    float* outVel = outPos + (size_t)n * 3;

    // Workspace layout (~59.3 MB of f32):
    float* ws   = (float*)d_ws;
    float* grid = ws;                       // MESH3
    float* Are  = grid + MESH3;             // complex ping/pong triples
    float* Aim  = Are + MESH3;
    float* Bre  = Aim + MESH3;
    float* Bim  = Bre + MESH3;
    float* Cre  = Bim + MESH3;
    float* Cim  = Cre + MESH3;
    float* Mfr  = Cim + MESH3;              // forward twiddles (128x128)
    float* Mfi  = Mfr + NLINES;
    float* Mir  = Mfi + NLINES;             // inverse twiddles (with 1/128)
    float* Mii  = Mir + NLINES;

    const int THR = 256;
    const int fftBlocks  = (NLINES / 16 / 8) * (MESH / 16);  // 1024 blocks
    const int meshBlocks = (MESH3 + THR - 1) / THR;
    const int partBlocks = (n + THR - 1) / THR;

    // 1. density grid = 0 ; twiddle matrices
    pgd_zero<<<meshBlocks, THR, 0, stream>>>(grid, MESH3);
    pgd_twiddles<<<(NLINES + THR - 1) / THR, THR, 0, stream>>>(Mfr, Mfi, Mir, Mii);

    // 2. CIC paint
    pgd_paint<<<partBlocks, THR, 0, stream>>>(pos, grid, n);

    // 3. forward 3D FFT (3 rotating WMMA-GEMM passes); pass 3 fuses the
    //    PGD band-pass * inverse-Laplacian filter -> pot_k in A.
    pgd_fft_pass<false, true, false, false><<<fftBlocks, THR, 0, stream>>>(
        grid, nullptr, Mfr, Mfi, Are, Aim, 0);
    pgd_fft_pass<true, true, false, false><<<fftBlocks, THR, 0, stream>>>(
        Are, Aim, Mfr, Mfi, Bre, Bim, 0);
    pgd_fft_pass<true, true, true, false><<<fftBlocks, THR, 0, stream>>>(
        Bre, Bim, Mfr, Mfi, Are, Aim, 0);

    // 4. per-axis: gradient fused into inverse pass 1; pass 3 writes the
    //    real-space force (real part only); then CIC gather + boost.
    for (int axis = 0; axis < 3; ++axis) {
        pgd_fft_pass<true, true, false, true><<<fftBlocks, THR, 0, stream>>>(
            Are, Aim, Mir, Mii, Bre, Bim, axis);
        pgd_fft_pass<true, true, false, false><<<fftBlocks, THR, 0, stream>>>(
            Bre, Bim, Mir, Mii, Cre, Cim, 0);
        pgd_fft_pass<true, false, false, false><<<fftBlocks, THR, 0, stream>>>(
            Cre, Cim, Mir, Mii, Bre, nullptr, 0);
        pgd_read<<<partBlocks, THR, 0, stream>>>(Bre, pos, vel, fdf, outVel, axis, n);
    }

    // 5. pos passthrough (output tuple element 0)
    (void)hipMemcpyAsync(outPos, pos, (size_t)n * 3 * sizeof(float),
                         hipMemcpyDeviceToDevice, stream);
}